// TensorProductBlock_18245021073788
// MI455X (gfx1250) — compile-verified
//
#include <hip/hip_runtime.h>
#include <math.h>

typedef float v2f __attribute__((ext_vector_type(2)));
typedef float v8f __attribute__((ext_vector_type(8)));

#define DD    40
#define SP    64000          // 40^3
#define NB    4
#define NPTS  (NB * SP)      // 256000
#define CIN   36
#define COUT  36
#define KVOL  27
#define KDIM  (CIN * KVOL)   // 972
#define EPSV  1e-5f

// ---------------- workspace layout (bytes) ----------------
// [0, 64)                     : stats accumulators (10 floats)
// [256, 256 + 48*972*4)       : Kmat [48][972]  (rows 36..47 zero)
// [186880, +4*64000*36*4)     : xnorm, channel-last [b][sp][36]
#define WS_KMAT_OFF   256
#define WS_XN_OFF     186880

struct KPtrs { const float* w[12]; const float* ba[12]; };

// ---------------------------------------------------------------------------
__global__ void k_zero_stats(float* s) {
  if (threadIdx.x < 16) s[threadIdx.x] = 0.f;
}

// ---------------------------------------------------------------------------
// Pass 1: per-multiplicity sums / sums-of-squares over batch+spatial.
__global__ __launch_bounds__(256) void k_stats(const float* __restrict__ sv5,
                                               float* __restrict__ st) {
  float a[10];
#pragma unroll
  for (int i = 0; i < 10; ++i) a[i] = 0.f;
  int p = blockIdx.x * 256 + threadIdx.x;
  if (p < NPTS) {
    int b = p / SP; int sp = p - b * SP;
    const float* in = sv5 + (size_t)b * 18 * SP + sp;
    float s0 = in[0], s1 = in[(size_t)SP];
    float v00 = in[(size_t)2*SP], v01 = in[(size_t)3*SP], v02 = in[(size_t)4*SP];
    float v10 = in[(size_t)5*SP], v11 = in[(size_t)6*SP], v12 = in[(size_t)7*SP];
    float l0 = 0.f, l1 = 0.f;
#pragma unroll
    for (int i = 0; i < 5; ++i) { float q = in[(size_t)(8+i)*SP];  l0 += q*q; }
#pragma unroll
    for (int i = 0; i < 5; ++i) { float q = in[(size_t)(13+i)*SP]; l1 += q*q; }
    float n0 = v00*v00 + v01*v01 + v02*v02;
    float n1 = v10*v10 + v11*v11 + v12*v12;
    a[0]=s0; a[1]=s1; a[2]=s0*s0; a[3]=s1*s1;
    a[4]=n0; a[5]=n1; a[6]=l0; a[7]=l1; a[8]=n0*n0; a[9]=n1*n1;
  }
#pragma unroll
  for (int off = 16; off > 0; off >>= 1) {
#pragma unroll
    for (int i = 0; i < 10; ++i) a[i] += __shfl_down(a[i], off);
  }
  __shared__ float blk[10];
  if (threadIdx.x < 10) blk[threadIdx.x] = 0.f;
  __syncthreads();
  if ((threadIdx.x & 31) == 0) {
#pragma unroll
    for (int i = 0; i < 10; ++i) atomicAdd(&blk[i], a[i]);
  }
  __syncthreads();
  if (threadIdx.x < 10) atomicAdd(&st[threadIdx.x], blk[threadIdx.x]);
}

// ---------------------------------------------------------------------------
// Pass 2: tensor product (raw v x v) + SE3 batchnorm, write channel-last xnorm.
__global__ __launch_bounds__(256) void k_norm(const float* __restrict__ sv5,
                                              const float* __restrict__ st,
                                              float* __restrict__ xn) {
  int p = blockIdx.x * 256 + threadIdx.x;
  if (p >= NPTS) return;
  int b = p / SP; int sp = p - b * SP;
  const float* in = sv5 + (size_t)b * 18 * SP + sp;
  const float Ninv = 1.f / (float)NPTS;
  float mean0 = st[0] * Ninv, mean1 = st[1] * Ninv;
  float inv_s0 = rsqrtf(st[2]*Ninv - mean0*mean0 + EPSV);
  float inv_s1 = rsqrtf(st[3]*Ninv - mean1*mean1 + EPSV);
  float inv_v0 = rsqrtf(st[4]*Ninv * (1.f/3.f) + EPSV);
  float inv_v1 = rsqrtf(st[5]*Ninv * (1.f/3.f) + EPSV);
  float inv_l0 = rsqrtf(st[6]*Ninv * (1.f/5.f) + EPSV);
  float inv_l1 = rsqrtf(st[7]*Ninv * (1.f/5.f) + EPSV);
  float inv_t0 = rsqrtf(st[8]*Ninv * (1.f/9.f) + EPSV);
  float inv_t1 = rsqrtf(st[9]*Ninv * (1.f/9.f) + EPSV);

  float* o = xn + (size_t)p * CIN;
  o[0] = (in[0] - mean0) * inv_s0;
  o[1] = (in[(size_t)SP] - mean1) * inv_s1;
  float v0[3], v1[3];
#pragma unroll
  for (int i = 0; i < 3; ++i) { v0[i] = in[(size_t)(2+i)*SP]; v1[i] = in[(size_t)(5+i)*SP]; }
#pragma unroll
  for (int i = 0; i < 3; ++i) { o[2+i] = v0[i]*inv_v0; o[5+i] = v1[i]*inv_v1; }
#pragma unroll
  for (int i = 0; i < 5; ++i) { o[8+i]  = in[(size_t)(8+i)*SP]  * inv_l0;
                                o[13+i] = in[(size_t)(13+i)*SP] * inv_l1; }
#pragma unroll
  for (int i = 0; i < 3; ++i)
#pragma unroll
    for (int j = 0; j < 3; ++j) {
      o[18 + i*3 + j] = v0[i]*v0[j]*inv_t0;
      o[27 + i*3 + j] = v1[i]*v1[j]*inv_t1;
    }
}

// ---------------------------------------------------------------------------
// Assemble Kmat[48][972]: Kmat[row][r*36+c] = sum_n w[idx](u,v,n)*basis[idx](n,io,j,r)
__global__ __launch_bounds__(256) void k_assemble(KPtrs kp, float* __restrict__ Kmat) {
  for (int e = blockIdx.x * blockDim.x + threadIdx.x; e < 48 * KDIM;
       e += gridDim.x * blockDim.x) {
    int row = e / KDIM; int kk = e - row * KDIM;
    int r = kk / CIN;  int c  = kk - r * CIN;
    float val = 0.f;
    if (row < COUT) {
      int bo, u, io, dof;
      if      (row < 4)  { bo = 0; dof = 1; u = row;          io = 0; }
      else if (row < 16) { bo = 1; dof = 3; u = (row-4)/3;    io = (row-4)%3; }
      else               { bo = 2; dof = 5; u = (row-16)/5;   io = (row-16)%5; }
      int bi, v, j, dif;
      if      (c < 2)    { bi = 0; dif = 1; v = c;            j = 0; }
      else if (c < 8)    { bi = 1; dif = 3; v = (c-2)/3;      j = (c-2)%3; }
      else if (c < 18)   { bi = 2; dif = 5; v = (c-8)/5;      j = (c-8)%5; }
      else               { bi = 3; dif = 9; v = (c-18)/9;     j = (c-18)%9; }
      int idx = bo * 4 + bi;
      const float* w  = kp.w[idx];
      const float* ba = kp.ba[idx];
#pragma unroll
      for (int n = 0; n < 4; ++n)
        val += w[(u*2 + v)*4 + n] * ba[((size_t)((n*dof + io)*dif + j))*KVOL + r];
    }
    Kmat[e] = val;
  }
}

// ---------------------------------------------------------------------------
// Implicit-GEMM 3D conv with V_WMMA_F32_16X16X4_F32.
// Block: 128 thr (4 wave32) on an 8x4x4 output tile. Each wave owns z-slice
// lz = wave_id and TWO 16-point N-groups (ly 0..1 and ly 2..3), so per K-step
// it issues 3 A-loads + 2 B-loads + 6 WMMAs (A traffic amortized 2x).
// GEMM K-order k = r*36 + c keeps each 4-wide K-step channel-contiguous:
// B fragment = one ds_load_b64 from the channel-last LDS halo tile.
__global__ __launch_bounds__(128) void k_conv(const float* __restrict__ xnorm,
                                              const float* __restrict__ Kmat,
                                              float* __restrict__ y) {
  __shared__ float xs[6 * 6 * 10 * CIN];   // halo tile, channel-last: 51840 B

  int id = blockIdx.x;
  int xt = id % 5; int yt = (id / 5) % 10; int zt = (id / 50) % 10; int b = id / 500;
  int x0 = xt * 8, y0 = yt * 4, z0 = zt * 4;

  // Cooperative halo load (zero-fill at pad borders)
  for (int idx = threadIdx.x; idx < 6 * 6 * 10 * CIN; idx += 128) {
    int c = idx % CIN; int s = idx / CIN;
    int xx = s % 10; int s2 = s / 10; int yy = s2 % 6; int zz = s2 / 6;
    int gx = x0 + xx - 1, gy = y0 + yy - 1, gz = z0 + zz - 1;
    float v = 0.f;
    if (gx >= 0 && gx < DD && gy >= 0 && gy < DD && gz >= 0 && gz < DD)
      v = xnorm[((size_t)(b * SP + (gz * DD + gy) * DD + gx)) * CIN + c];
    xs[idx] = v;
  }
  __syncthreads();

  int lane = threadIdx.x & 31;
  int w    = threadIdx.x >> 5;   // z-slice
  int half = lane >> 4;          // fragment lane-half (K = k0 + 2*half + {0,1})
  int n16  = lane & 15;          // N column / A row within tile
  int lx = n16 & 7;
  int ly0 = n16 >> 3;            // group 0: ly in {0,1};  group 1: ly0+2

  // per-lane LDS element base for B fragments (channel-last tile)
  int sbase = ((w * 6 + ly0) * 10 + lx) * CIN + 2 * half;   // group 0
  // group 1 is +2 rows in y: +2*10*36 = +720 elements (immediate-foldable)
  const float* Ab = Kmat + n16 * KDIM + 2 * half;           // A row base

  v8f a00 = {}, a01 = {}, a02 = {};   // N-group 0, m-tiles 0..2
  v8f a10 = {}, a11 = {}, a12 = {};   // N-group 1

  for (int r = 0; r < KVOL; ++r) {
    int kz = r / 9; int rr = r - kz * 9; int ky = rr / 3; int kx = rr - ky * 3;
    int koff = ((kz * 6 + ky) * 10 + kx) * CIN;             // wave-uniform
    const float* Ar = Ab + r * CIN;
    const float* Bp = xs + sbase + koff;
#pragma unroll
    for (int c0 = 0; c0 < CIN; c0 += 4) {
      v2f b0 = *(const v2f*)(Bp + c0);                      // ds_load_b64
      v2f b1 = *(const v2f*)(Bp + 720 + c0);                // ds_load_b64
      v2f f0 = *(const v2f*)(Ar + c0);                      // global_load_b64
      v2f f1 = *(const v2f*)(Ar + 16 * KDIM + c0);
      v2f f2 = *(const v2f*)(Ar + 32 * KDIM + c0);
      a00 = __builtin_amdgcn_wmma_f32_16x16x4_f32(false, f0, false, b0, (short)0, a00, false, false);
      a10 = __builtin_amdgcn_wmma_f32_16x16x4_f32(false, f0, false, b1, (short)0, a10, false, false);
      a01 = __builtin_amdgcn_wmma_f32_16x16x4_f32(false, f1, false, b0, (short)0, a01, false, false);
      a11 = __builtin_amdgcn_wmma_f32_16x16x4_f32(false, f1, false, b1, (short)0, a11, false, false);
      a02 = __builtin_amdgcn_wmma_f32_16x16x4_f32(false, f2, false, b0, (short)0, a02, false, false);
      a12 = __builtin_amdgcn_wmma_f32_16x16x4_f32(false, f2, false, b1, (short)0, a12, false, false);
    }
  }

  // Store. C/D layout: VGPR v, lane -> M = v + 8*half, N = lane&15.
  // group g point: (lx, g*2 + ly0, w)
  int gx = x0 + lx, gz = z0 + w;
  float* ybase = y + (size_t)b * COUT * SP;
#pragma unroll
  for (int g = 0; g < 2; ++g) {
    int gy = y0 + g * 2 + ly0;
    float* yb = ybase + (size_t)((gz * DD + gy) * DD + gx);
    v8f c0 = g ? a10 : a00;
    v8f c1 = g ? a11 : a01;
    v8f c2 = g ? a12 : a02;
#pragma unroll
    for (int v = 0; v < 8; ++v) {
      int m0 = v + 8 * half;
      float o0 = c0[v];
      if (m0 < 4) o0 = fmaxf(o0, 0.f);                      // ReLU on scalar outputs
      yb[(size_t)m0 * SP]        = o0;
      yb[(size_t)(16 + m0) * SP] = c1[v];
      if (m0 < 4) yb[(size_t)(32 + m0) * SP] = c2[v];       // rows 36..47 are padding
    }
  }
}

// ---------------------------------------------------------------------------
extern "C" void kernel_launch(void* const* d_in, const int* in_sizes, int n_in,
                              void* d_out, int out_size, void* d_ws, size_t ws_size,
                              hipStream_t stream) {
  const float* sv5 = (const float*)d_in[0];

  static const int DOv[3] = {1, 3, 5};
  static const int DIv[4] = {1, 3, 5, 9};
  KPtrs kp;
  if (n_in >= 25) {
    for (int i = 0; i < 12; ++i) {
      kp.w[i]  = (const float*)d_in[1 + i];
      kp.ba[i] = (const float*)d_in[13 + i];
    }
  } else {
    // lists passed as two concatenated blobs
    const float* wb = (const float*)d_in[1];
    const float* bb = (const float*)d_in[2];
    size_t wo = 0, bo = 0;
    for (int idx = 0; idx < 12; ++idx) {
      kp.w[idx]  = wb + wo; wo += 32;                       // 4*2*4
      kp.ba[idx] = bb + bo; bo += (size_t)4 * DOv[idx/4] * DIv[idx%4] * KVOL;
    }
  }

  float* stats = (float*)d_ws;
  float* Kmat  = (float*)((char*)d_ws + WS_KMAT_OFF);
  float* xnorm = (float*)((char*)d_ws + WS_XN_OFF);
  float* out   = (float*)d_out;

  k_zero_stats<<<1, 32, 0, stream>>>(stats);
  k_stats   <<<NPTS / 256, 256, 0, stream>>>(sv5, stats);
  k_norm    <<<NPTS / 256, 256, 0, stream>>>(sv5, stats, xnorm);
  k_assemble<<<64, 256, 0, stream>>>(kp, Kmat);
  k_conv    <<<2000, 128, 0, stream>>>(xnorm, Kmat, out);
}